// Conv2d_mvm_5428838662439
// MI455X (gfx1250) — compile-verified
//
#include <hip/hip_runtime.h>

typedef float v2f __attribute__((ext_vector_type(2)));
typedef float v8f __attribute__((ext_vector_type(8)));
typedef v2f v2f_u __attribute__((aligned(4)));   // 4B-aligned b64 load (DWORD mode OK)

#define C_IN    8
#define C_OUT   8
#define H_      2048
#define W_      2048
#define H_OUT   2046
#define W_OUT   2046
#define KTOT    72            // C_IN * 3 * 3
#define NCH     24            // chunks: one per (ic,kh); K padded 3->4 (kw=3 zero weight)
#define TPR     128           // 16-px tiles per output row
#define HTRI    682           // 2046 / 3 row-triples
#define NTRIPLES (HTRI * TPR)
#define NMAX    (C_IN * H_ * W_ - 1)

// One 16-wide x 3-tall output tile. CLAMP instantiation used only for the single
// final triple (t == NTRIPLES-1), the only place a b64 load can run past the buffer.
template<bool CLAMP>
__device__ __forceinline__ void conv_triple(const float* __restrict__ x,
                                            float* __restrict__ out,
                                            const v2f (&bfrag)[NCH],
                                            int hb, int wb, int lo, int hi)
{
    // lane's b64 base: covers x[.][row][wb+lo+2*hi, +1]
    const int pb = hb * W_ + wb + lo + 2 * hi;

    v8f acc0 = {}, acc1 = {}, acc2 = {};
    #pragma unroll
    for (int ic = 0; ic < C_IN; ++ic) {
        const int base = pb + ic * (H_ * W_);    // 32-bit index; r*W folds into imm24
        // 5 input rows feed 9 WMMAs (rows hb..hb+4); one b64 load per row
        v2f F[5];
        #pragma unroll
        for (int r = 0; r < 5; ++r) {
            int idx = base + r * W_;
            if (CLAMP) idx = min(idx, NMAX - 1);
            F[r] = *(const v2f_u*)(x + idx);
        }
        #pragma unroll
        for (int kh = 0; kh < 3; ++kh) {
            const v2f b = bfrag[ic * 3 + kh];
            acc0 = __builtin_amdgcn_wmma_f32_16x16x4_f32(
                       false, F[kh + 0], false, b, (short)0, acc0, false, false);
            acc1 = __builtin_amdgcn_wmma_f32_16x16x4_f32(
                       false, F[kh + 1], false, b, (short)0, acc1, false, false);
            acc2 = __builtin_amdgcn_wmma_f32_16x16x4_f32(
                       false, F[kh + 2], false, b, (short)0, acc2, false, false);
        }
    }

    // D layout: VGPR r, lane L -> out[oc = L&15][row][wb + 8*(L>=16) + r]
    if (lo < C_OUT) {
        const int wcol = wb + hi * 8;
        const size_t base = (size_t)lo * ((size_t)H_OUT * W_OUT)
                          + (size_t)hb * W_OUT + wcol;
        if (wb + 16 <= W_OUT) {
            #pragma unroll
            for (int r = 0; r < 8; ++r) {
                out[base + r]             = acc0[r];
                out[base + W_OUT + r]     = acc1[r];
                out[base + 2 * W_OUT + r] = acc2[r];
            }
        } else {                                // last column tile: width 14
            #pragma unroll
            for (int r = 0; r < 8; ++r) {
                if (wcol + r < W_OUT) {
                    out[base + r]             = acc0[r];
                    out[base + W_OUT + r]     = acc1[r];
                    out[base + 2 * W_OUT + r] = acc2[r];
                }
            }
        }
    }
}

__global__ __launch_bounds__(256) void conv3x3_wmma_f32(
        const float* __restrict__ x,
        const float* __restrict__ wgt,
        float* __restrict__ out)
{
    const int lane = threadIdx.x & 31;
    const int hi   = lane >> 4;            // 0: lanes 0-15 (K slots 0,1), 1: lanes 16-31 (slots 2,3)
    const int lo   = lane & 15;
    const int gw   = (int)((blockIdx.x * blockDim.x + threadIdx.x) >> 5);
    const int nw   = (int)((gridDim.x * blockDim.x) >> 5);

    // ---- B fragments: 24 chunks (ic,kh), resident in 48 VGPRs ----
    // chunk K-slot s corresponds to kw = s; kw==3 is phantom (zero weight).
    // lane half gives slots {0,1} or {2,3}; column n = lo = output channel.
    v2f bfrag[NCH];
    #pragma unroll
    for (int ic = 0; ic < C_IN; ++ic) {
        #pragma unroll
        for (int kh = 0; kh < 3; ++kh) {
            const int c   = ic * 3 + kh;
            const int kw0 = 2 * hi;          // slot kw for .x
            float b0 = 0.f, b1 = 0.f;
            if (lo < C_OUT) {
                const int base = lo * KTOT + ic * 9 + kh * 3;
                b0 = wgt[base + kw0];
                if (kw0 + 1 < 3) b1 = wgt[base + kw0 + 1];   // kw==3 -> 0
            }
            bfrag[c].x = b0;
            bfrag[c].y = b1;
        }
    }

    for (int t = gw; t < NTRIPLES; t += nw) {
        const int hb = (t >> 7) * 3;            // first of 3 output rows
        const int wb = (t & (TPR - 1)) << 4;    // first output column

        // prefetch next triple's first row (global_prefetch_b8)
        {
            const int tn = t + nw;
            if (tn < NTRIPLES) {
                const int pn = ((tn >> 7) * 3) * W_ + ((tn & (TPR - 1)) << 4) + lo;
                __builtin_prefetch(x + pn, 0, 3);
            }
        }

        if (t < NTRIPLES - 1) {
            conv_triple<false>(x, out, bfrag, hb, wb, lo, hi);  // fast: no clamps
        } else {
            conv_triple<true >(x, out, bfrag, hb, wb, lo, hi);  // sole edge triple
        }
    }
}

extern "C" void kernel_launch(void* const* d_in, const int* in_sizes, int n_in,
                              void* d_out, int out_size, void* d_ws, size_t ws_size,
                              hipStream_t stream) {
    (void)in_sizes; (void)n_in; (void)out_size; (void)d_ws; (void)ws_size;
    const float* x   = (const float*)d_in[0];
    const float* wgt = (const float*)d_in[1];
    float*       out = (float*)d_out;

    // 1024 blocks x 256 threads = 8192 waves; 87,296 row-triples -> ~10.7/wave
    hipLaunchKernelGGL(conv3x3_wmma_f32, dim3(1024), dim3(256), 0, stream,
                       x, wgt, out);
}